// E52QuadraticGateCell_45792941310477
// MI455X (gfx1250) — compile-verified
//
#include <hip/hip_runtime.h>

// Problem dims
#define TT 1024
#define BB 8
#define DD 1024
#define BD (BB * DD)        // 8192 floats per h timestep
#define NWG 16              // workgroups in the recurrence kernel
#define RT 512              // recurrence kernel threads (16 waves)
#define ESL 64              // e-columns owned per workgroup (16*64 = 1024)
#define PITCH 1028          // LDS row pitch (floats): conflict-free wmma frag loads
#define SPECTRAL_RADIUS 0.999f
#define EPSF 1e-8f

typedef float v2f __attribute__((ext_vector_type(2)));
typedef float v8f __attribute__((ext_vector_type(8)));

// ---------------------------------------------------------------------------
// CDNA5 async memory->LDS copy (ASYNCcnt-tracked, no VGPR data movement).
// LDS address operand = raw LDS byte offset = low 32 bits of generic pointer.
// ---------------------------------------------------------------------------
__device__ __forceinline__ void async_load_b128_to_lds(const void* gaddr,
                                                       void* lds_ptr) {
  unsigned lds_off = (unsigned)(size_t)lds_ptr;   // generic->AS3 truncate
  asm volatile("global_load_async_to_lds_b128 %0, %1, off"
               :: "v"(lds_off), "v"(gaddr)
               : "memory");
}
__device__ __forceinline__ void wait_async0() {
  asm volatile("s_wait_asynccnt 0x0" ::: "memory");
}

// ---------------------------------------------------------------------------
// Block-wide sum reduction (blockDim.x == 1024)
// ---------------------------------------------------------------------------
__device__ inline float block_reduce_sum(float x, float* buf) {
  const int tid = threadIdx.x;
  buf[tid] = x;
  __syncthreads();
  for (int s = 512; s > 0; s >>= 1) {
    if (tid < s) buf[tid] += buf[tid + s];
    __syncthreads();
  }
  float r = buf[0];
  __syncthreads();
  return r;
}

// ---------------------------------------------------------------------------
// Kernel A: 3-step power iteration -> scale; zero step-barrier counter; seed
// h[0] = h0 in the output h-region. One block, 1024 threads.
// ---------------------------------------------------------------------------
__global__ __launch_bounds__(1024) void spectral_init_kernel(
    const float* __restrict__ W, const float* __restrict__ u_in,
    const float* __restrict__ h0, float* __restrict__ scale_out,
    unsigned int* __restrict__ counter, float* __restrict__ h_out) {
  __shared__ float su[DD];
  __shared__ float sv[DD];
  __shared__ float rbuf[DD];
  const int tid = threadIdx.x;

  if (tid == 0) *counter = 0u;                  // reset per-launch (determinism)
  su[tid] = u_in[tid];
  for (int i = tid; i < BD; i += DD) h_out[i] = h0[i];   // h[0] = h0
  __syncthreads();

  float sigma = 0.f;
  for (int it = 0; it < 3; ++it) {
    float acc = 0.f;                             // v = normalize(W^T u)
    for (int i = 0; i < DD; ++i) acc = fmaf(W[(size_t)i * DD + tid], su[i], acc);
    float nv = sqrtf(block_reduce_sum(acc * acc, rbuf));
    sv[tid] = acc / (nv + EPSF);
    __syncthreads();
    float acc2 = 0.f;                            // u = normalize(W v)
    const float* row = W + (size_t)tid * DD;
    for (int j = 0; j < DD; ++j) acc2 = fmaf(row[j], sv[j], acc2);
    float nu = sqrtf(block_reduce_sum(acc2 * acc2, rbuf));
    su[tid] = acc2 / (nu + EPSF);
    __syncthreads();
    sigma = nu * nu / (nu + EPSF);               // = u.(Wv), exact ref math
  }
  if (tid == 0) *scale_out = SPECTRAL_RADIUS / (fabsf(sigma) + EPSF);
}

// ---------------------------------------------------------------------------
// Kernel B: WxB[m,e] = scale * sum_d x[m,d]*W[e,d] + b[e]
// M=8192, N=1024, K=1024; fp32 WMMA 16x16x4. Block = one e-tile (16 e) whose
// 16 W rows are async-staged once into LDS (shared by all 8 waves); each wave
// streams one 16-row x tile from global. grid = (64 e-tiles, 64 m-groups).
// ---------------------------------------------------------------------------
__global__ __launch_bounds__(256) void wx_gemm_kernel(
    const float* __restrict__ x, const float* __restrict__ W,
    const float* __restrict__ bvec, const float* __restrict__ scale_ptr,
    float* __restrict__ wxb) {
  __shared__ float sB[16 * PITCH];                // 65792 B
  const float scale = *scale_ptr;
  const int tid = threadIdx.x;
  const int wave = tid >> 5, lane = tid & 31;
  const int l15 = lane & 15, hi = lane >> 4;
  const int mtile = blockIdx.y * 8 + wave;        // 0..511
  const int e0 = blockIdx.x * 16;

  // Async-stage W[e0..e0+15][:] into LDS with pitch-1028 rows.
  for (int idx = tid; idx < 16 * (DD / 4); idx += 256) {
    const int r = idx >> 8;                       // 256 b128 per row
    const int c4 = (idx & 255) * 4;
    async_load_b128_to_lds(W + (size_t)(e0 + r) * DD + c4, sB + r * PITCH + c4);
  }
  wait_async0();
  __syncthreads();

  const float* xrow = x + (size_t)(mtile * 16 + l15) * DD + 2 * hi;  // A frag
  const float* brow = sB + l15 * PITCH + 2 * hi;                     // B frag

  v8f c = {};
  for (int d0 = 0; d0 < DD; d0 += 4) {
    v2f a = *(const v2f*)(xrow + d0);
    v2f b2 = *(const v2f*)(brow + d0);
    c = __builtin_amdgcn_wmma_f32_16x16x4_f32(false, a, false, b2,
                                              (short)0, c, false, false);
  }

  const int col = e0 + l15;
  const float bb = bvec[col];
#pragma unroll
  for (int r = 0; r < 8; ++r) {
    const int m = mtile * 16 + r + 8 * hi;        // C layout: VGPR r -> r+8*half
    wxb[(size_t)m * DD + col] = c[r] * scale + bb;
  }
}

// ---------------------------------------------------------------------------
// Kernel C: sequential scan h_t = WxB_t + scale*(h_{t-1} @ W_slice^T).
// 16 persistent workgroups x 512 threads (16 waves). Each WG owns 64
// e-columns of W, async-staged once into LDS. Per step, h[t] (32 KB) is
// async-staged from L2. 16 waves = 4 e-tiles x 4-way K split -> only 64
// dependent WMMAs on the critical path; partials combined via LDS. One
// device-scope monotonic-counter barrier per step; h region of d_out is the
// exchange buffer (step t reads h[t], writes h[t+1] -- no WAR hazard).
// ---------------------------------------------------------------------------
__global__ __launch_bounds__(RT) void recurrence_kernel(
    const float* __restrict__ W, const float* __restrict__ scale_ptr,
    const float* __restrict__ wxb, unsigned int* counter,
    float* __restrict__ h_out, float* __restrict__ out_sq) {
  __shared__ float sW[ESL * PITCH];     // 263168 B : We slice (unscaled)
  __shared__ float sH[BB * PITCH];      //  32896 B : staged h_{t-1}
  __shared__ float sRed[4 * 3 * 256];   //  12288 B : 3 K-partials per e-tile

  const int tid = threadIdx.x;
  const int wg = blockIdx.x;
  const int e_base = wg * ESL;
  const float scale = *scale_ptr;

  // One-time async fill of our 64 W rows (pure copy; scale applied at epilogue)
  for (int idx = tid; idx < ESL * (DD / 4); idx += RT) {
    const int r = idx >> 8;
    const int c4 = (idx & 255) * 4;
    async_load_b128_to_lds(W + (size_t)(e_base + r) * DD + c4,
                           sW + r * PITCH + c4);
  }
  wait_async0();

  const int wave = tid >> 5, lane = tid & 31;
  const int l15 = lane & 15, hi = lane >> 4;
  const int et = wave >> 2;             // e-tile 0..3
  const int kq = wave & 3;              // K quarter
  const int d_base = kq * 256 + 2 * hi;
  const bool avalid = (l15 < 8);        // batch rows 8..15 are zero padding
  const v2f azero = {0.f, 0.f};

  for (int t = 0; t < TT; ++t) {
    // Async-stage h[t] (32 KB, 2048 b128) from L2 into LDS.
    const float* hsrc = h_out + (size_t)t * BD;
    for (int idx = tid; idx < BD / 4; idx += RT) {
      const int r = idx >> 8;
      const int c4 = (idx & 255) * 4;
      async_load_b128_to_lds(hsrc + (size_t)r * DD + c4, sH + r * PITCH + c4);
    }
    wait_async0();
    __syncthreads();

    // Prefetch next step's WxB slice for this WG (global_prefetch_b8).
    if (t + 1 < TT && tid < 64) {
      const float* p = wxb + (size_t)((t + 1) * BB + (tid >> 3)) * DD +
                       e_base + (tid & 7) * 8;
      __builtin_prefetch(p, 0, 0);
    }

    // Accumulate C += A(16x4) x B(4x16) over this wave's K quarter (64 wmma).
    v8f c = {};
    const float* arow = sH + l15 * PITCH + d_base;
    const float* brow = sW + (et * 16 + l15) * PITCH + d_base;
#pragma unroll 4
    for (int kk = 0; kk < 64; ++kk) {
      v2f a = avalid ? *(const v2f*)(arow + kk * 4) : azero;
      v2f b2 = *(const v2f*)(brow + kk * 4);
      c = __builtin_amdgcn_wmma_f32_16x16x4_f32(false, a, false, b2,
                                                (short)0, c, false, false);
    }

    // Combine the 4 K-quarters via LDS.
    if (kq != 0) {
#pragma unroll
      for (int r = 0; r < 8; ++r)
        sRed[(et * 3 + (kq - 1)) * 256 + r * 32 + lane] = c[r];
    }
    __syncthreads();
    if (kq == 0) {
#pragma unroll
      for (int p = 0; p < 3; ++p)
#pragma unroll
        for (int r = 0; r < 8; ++r)
          c[r] += sRed[(et * 3 + p) * 256 + r * 32 + lane];
      if (hi == 0) {  // lanes 0..15 hold valid batch rows m = r (half 0)
        const int col = e_base + et * 16 + l15;
#pragma unroll
        for (int r = 0; r < 8; ++r) {
          const float val = c[r] * scale + wxb[(size_t)(t * BB + r) * DD + col];
          h_out[(size_t)(t + 1) * BD + (size_t)r * DD + col] = val;
          out_sq[(size_t)t * BD + (size_t)r * DD + col] = val * val;
        }
      }
    }
    __syncthreads();

    // Device-scope step barrier: monotonic counter, 16 signals per step.
    if (tid == 0) {
      __threadfence();
      __hip_atomic_fetch_add(counter, 1u, __ATOMIC_RELEASE,
                             __HIP_MEMORY_SCOPE_AGENT);
      const unsigned target = (unsigned)(NWG * (t + 1));
      while (__hip_atomic_load(counter, __ATOMIC_ACQUIRE,
                               __HIP_MEMORY_SCOPE_AGENT) < target) {
        __builtin_amdgcn_s_sleep(4);
      }
    }
    __syncthreads();
  }
}

// ---------------------------------------------------------------------------
// Launch: inputs (dict order): x[T,B,D], h0[B,D], W[D,D], b[D], u[D], all f32.
// d_out = output[T,B,D] ++ h[T+1,B,D] (f32).
// Workspace: [0]=scale, [256B]=counter, [512B..]=WxB (T*B*D f32, 32 MB).
// ---------------------------------------------------------------------------
extern "C" void kernel_launch(void* const* d_in, const int* in_sizes, int n_in,
                              void* d_out, int out_size, void* d_ws, size_t ws_size,
                              hipStream_t stream) {
  const float* x  = (const float*)d_in[0];
  const float* h0 = (const float*)d_in[1];
  const float* W  = (const float*)d_in[2];
  const float* b  = (const float*)d_in[3];
  const float* u  = (const float*)d_in[4];

  float* out_sq = (float*)d_out;                        // [T,B,D]
  float* h_out  = out_sq + (size_t)TT * BD;             // [T+1,B,D]

  float*        scale_ptr = (float*)d_ws;
  unsigned int* counter   = (unsigned int*)((char*)d_ws + 256);
  float*        wxb       = (float*)((char*)d_ws + 512);

  spectral_init_kernel<<<1, 1024, 0, stream>>>(W, u, h0, scale_ptr, counter, h_out);
  wx_gemm_kernel<<<dim3(64, 64), 256, 0, stream>>>(x, W, b, scale_ptr, wxb);
  recurrence_kernel<<<NWG, RT, 0, stream>>>(W, scale_ptr, wxb, counter,
                                            h_out, out_sq);
}